// GeneWiseDecoder_80891414052868
// MI455X (gfx1250) — compile-verified
//
#include <hip/hip_runtime.h>
#include <hip/hip_bf16.h>
#include <math.h>

// ---------------------------------------------------------------------------
// GeneWiseDecoder: 1000 independent per-gene MLPs over a shared batch.
//   h1 = gelu(x @ W1[g] + b1[g])      [128,128]x[128,512]
//   h2 = gelu(h1 @ W2[g] + b2[g])     [128,512]x[512,512]
//   out[:,g] = h2 @ W3[g] + b3[g]     [128,512]x[512]
// Memory-bound on fp32 weight streaming (~1.3 GB @ 23.3 TB/s ≈ 56 us floor),
// so we use exact-precision V_WMMA_F32_16X16X4_F32 (no benefit to low
// precision: weight bytes dominate and stay fp32). Weights are single-use ->
// non-temporal loads so the stream doesn't evict the L2-resident shared x.
// ---------------------------------------------------------------------------

#define N_GENES 1000
#define N_LATENT 128
#define N_HID 512
#define N_BATCH 128
#define H1_STRIDE 514   // 512 + 2 pad: stride%64==2 -> conflict-free b64 LDS reads, 8B aligned

typedef __attribute__((ext_vector_type(2))) float v2f;
typedef __attribute__((ext_vector_type(8))) float v8f;

__device__ __forceinline__ v8f wmma4(v2f a, v2f b, v8f c) {
  // (neg_a, A, neg_b, B, c_mod, C, reuse_a, reuse_b)
  return __builtin_amdgcn_wmma_f32_16x16x4_f32(false, a, false, b, (short)0, c,
                                               false, false);
}

__device__ __forceinline__ float gelu_exact(float v) {
  return 0.5f * v * (1.0f + erff(v * 0.70710678118654752f));
}

__global__ void __launch_bounds__(256)
gene_mlp_kernel(const float* __restrict__ x,
                const float* __restrict__ W1, const float* __restrict__ b1,
                const float* __restrict__ W2, const float* __restrict__ b2,
                const float* __restrict__ W3, const float* __restrict__ b3,
                float* __restrict__ out) {
  extern __shared__ float smem[];
  float* h1 = smem;                                 // [128][H1_STRIDE]
  float* oacc = smem + N_BATCH * H1_STRIDE;         // [128]

  const int g    = blockIdx.x;
  const int tid  = threadIdx.x;
  const int wave = tid >> 5;
  const int lane = tid & 31;
  const int lo   = lane & 15;      // column within 16-lane half
  const int hi   = lane >> 4;      // half-select
  const int kofs = hi * 2;         // A/B fragment K offset for this half

  const float* W1g = W1 + (size_t)g * N_LATENT * N_HID;
  const float* W2g = W2 + (size_t)g * N_HID * N_HID;
  const float* b1g = b1 + (size_t)g * N_HID;
  const float* b2g = b2 + (size_t)g * N_HID;
  const float* W3g = W3 + (size_t)g * N_HID;

  if (tid < N_BATCH) oacc[tid] = 0.0f;

  // ------------------------- Layer 1: x @ W1 -> h1 (LDS) -------------------
  for (int p = 0; p < 2; ++p) {
    const int n0 = wave * 64 + p * 32;   // 32-column slab: 2 N-tiles
    v8f c[2][8];
#pragma unroll
    for (int t = 0; t < 2; ++t)
#pragma unroll
      for (int m = 0; m < 8; ++m)
        c[t][m] = (v8f){0.f, 0.f, 0.f, 0.f, 0.f, 0.f, 0.f, 0.f};

    for (int k0 = 0; k0 < N_LATENT; k0 += 4) {
      v2f bfrag[2];
#pragma unroll
      for (int t = 0; t < 2; ++t) {
        // Single-use weight stream: non-temporal so it doesn't evict x in L2.
        const float* bp = W1g + (size_t)(k0 + kofs) * N_HID + n0 + 16 * t + lo;
        bfrag[t].x = __builtin_nontemporal_load(bp);
        bfrag[t].y = __builtin_nontemporal_load(bp + N_HID);
      }
#pragma unroll
      for (int m = 0; m < 8; ++m) {
        v2f a = *(const v2f*)(x + (size_t)(16 * m + lo) * N_LATENT + k0 + kofs);
        c[0][m] = wmma4(a, bfrag[0], c[0][m]);
        c[1][m] = wmma4(a, bfrag[1], c[1][m]);
      }
    }

    // epilogue: bias + exact GELU, scatter to LDS h1
#pragma unroll
    for (int t = 0; t < 2; ++t) {
      const int ncol = n0 + 16 * t + lo;
      const float bias = b1g[ncol];
#pragma unroll
      for (int m = 0; m < 8; ++m)
#pragma unroll
        for (int v = 0; v < 8; ++v) {
          const int row = 16 * m + v + 8 * hi;
          h1[row * H1_STRIDE + ncol] = gelu_exact(c[t][m][v] + bias);
        }
    }
  }
  __syncthreads();

  // --------------- Layer 2 + 3: h1 @ W2 -> gelu -> dot W3 ------------------
  for (int p = 0; p < 2; ++p) {
    const int n0 = wave * 64 + p * 32;
    v8f c[2][8];
#pragma unroll
    for (int t = 0; t < 2; ++t)
#pragma unroll
      for (int m = 0; m < 8; ++m)
        c[t][m] = (v8f){0.f, 0.f, 0.f, 0.f, 0.f, 0.f, 0.f, 0.f};

    for (int k0 = 0; k0 < N_HID; k0 += 4) {
      v2f bfrag[2];
#pragma unroll
      for (int t = 0; t < 2; ++t) {
        const float* bp = W2g + (size_t)(k0 + kofs) * N_HID + n0 + 16 * t + lo;
        bfrag[t].x = __builtin_nontemporal_load(bp);
        bfrag[t].y = __builtin_nontemporal_load(bp + N_HID);
      }
#pragma unroll
      for (int m = 0; m < 8; ++m) {
        v2f a = *(const v2f*)(h1 + (size_t)(16 * m + lo) * H1_STRIDE + k0 + kofs);
        c[0][m] = wmma4(a, bfrag[0], c[0][m]);
        c[1][m] = wmma4(a, bfrag[1], c[1][m]);
      }
    }

    // epilogue: bias + GELU, multiply by W3 column weight, reduce rows
#pragma unroll
    for (int t = 0; t < 2; ++t) {
      const int ncol = n0 + 16 * t + lo;
      const float bias = b2g[ncol];
      const float w3v = W3g[ncol];
#pragma unroll
      for (int m = 0; m < 8; ++m)
#pragma unroll
        for (int v = 0; v < 8; ++v) {
          float contrib = gelu_exact(c[t][m][v] + bias) * w3v;
          contrib += __shfl_xor(contrib, 1, 32);
          contrib += __shfl_xor(contrib, 2, 32);
          contrib += __shfl_xor(contrib, 4, 32);
          contrib += __shfl_xor(contrib, 8, 32);
          if (lo == 0) atomicAdd(&oacc[16 * m + v + 8 * hi], contrib);
        }
    }
  }
  __syncthreads();

  if (tid < N_BATCH) out[(size_t)tid * N_GENES + g] = oacc[tid] + b3[g];
}

extern "C" void kernel_launch(void* const* d_in, const int* in_sizes, int n_in,
                              void* d_out, int out_size, void* d_ws, size_t ws_size,
                              hipStream_t stream) {
  const float* x  = (const float*)d_in[0];
  const float* W1 = (const float*)d_in[1];
  const float* b1 = (const float*)d_in[2];
  const float* W2 = (const float*)d_in[3];
  const float* b2 = (const float*)d_in[4];
  const float* W3 = (const float*)d_in[5];
  const float* b3 = (const float*)d_in[6];
  float* out = (float*)d_out;

  const int smem_bytes = (N_BATCH * H1_STRIDE + N_BATCH) * (int)sizeof(float);
  (void)hipFuncSetAttribute((const void*)gene_mlp_kernel,
                            hipFuncAttributeMaxDynamicSharedMemorySize,
                            smem_bytes);
  gene_mlp_kernel<<<N_GENES, 256, smem_bytes, stream>>>(x, W1, b1, W2, b2, W3,
                                                        b3, out);
}